// DataWindowLoss_89790586290207
// MI455X (gfx1250) — compile-verified
//
#include <hip/hip_runtime.h>
#include <hip/hip_bf16.h>
#include <math.h>

// ---------------------------------------------------------------------------
// mean(|box5x5(x) - box5x5(y)|) = mean(|box5x5(x - y)|)   (linearity)
// x,y: [64,1,512,512] f32, conv padding=4 -> output [64,1,516,516]
// Bandwidth-bound: ~134 MB read @ 23.3 TB/s ~ 6 us (input is L2-resident).
// Vertical 5-tap done with V_WMMA_F32_16X16X4_F32 (banded W(16x20) x H(20x16)).
// Horizontal 5-tap folded into B-fragment construction from zero-padded d.
// ---------------------------------------------------------------------------

typedef float v2f __attribute__((ext_vector_type(2)));
typedef float v8f __attribute__((ext_vector_type(8)));

#define IMG_H   512
#define IMG_W   512
#define N_IMG   64
#define OUT_H   516
#define OUT_W   516
#define TILE_M  16                  // output rows per block
#define WIN_R   20                  // input rows needed: TILE_M + 4 halo
#define D_COLS  536                 // 4 left pad + 512 data + 20 right pad
#define N_CHUNK 33                  // 528 / 16 column chunks (512..527 padded)
#define ROW_TILES 33                // ceil(516/16)
#define NPART   (ROW_TILES * N_IMG) // 2112 partials
#define N_PADC  24                  // pad columns per row: 4 left + 20 right

__global__ __launch_bounds__(256)
void boxdiff_wmma_kernel(const float* __restrict__ x,
                         const float* __restrict__ y,
                         float* __restrict__ partials) {
    // dpad[wr][4+c] = x[row][c] - y[row][c]; zeros in halo rows and pad cols.
    __shared__ __align__(16) float dpad[WIN_R][D_COLS];
    __shared__ float red[256];

    const int tid     = threadIdx.x;
    const int img     = blockIdx.y;
    const int rowTile = blockIdx.x;
    const int oi0     = rowTile * TILE_M;
    const long imgBase = (long)img * IMG_H * IMG_W;

    // ---- Stage 1a: zero the pad columns (cols 0..3 and 516..535) ----------
    #pragma unroll
    for (int idx = tid; idx < WIN_R * N_PADC; idx += 256) {
        int wr = idx / N_PADC;
        int c  = idx % N_PADC;
        int col = (c < 4) ? c : (512 + c);   // 0..3 or 516..535
        dpad[wr][col] = 0.0f;
    }

    // ---- Stage 1b: all 20 rows of x/y in one load stream, d = x - y -------
    // 20 rows * 128 float4 quads = 2560 quad pairs; 256 threads * 10 iters.
    #pragma unroll
    for (int i = 0; i < 10; ++i) {
        const int q   = i * 256 + tid;       // global quad id
        const int wr  = q >> 7;              // window row 0..19
        const int qc  = q & 127;             // quad column 0..127
        const int rin = oi0 - 4 + wr;        // input row
        float4 d4 = make_float4(0.f, 0.f, 0.f, 0.f);
        if (rin >= 0 && rin < IMG_H) {
            const long off = imgBase + (long)rin * IMG_W + 4 * qc;
            float4 xv = *(const float4*)(x + off);
            float4 yv = *(const float4*)(y + off);
            d4 = make_float4(xv.x - yv.x, xv.y - yv.y, xv.z - yv.z, xv.w - yv.w);
        }
        *(float4*)(&dpad[wr][4 + 4 * qc]) = d4;
    }
    __syncthreads();

    // ---- Stage 2: vertical 5-tap as banded WMMA ---------------------------
    // out(16x16) = W(16x20) x H(20x16), K split into 5 chunks of 4.
    // H[k][n] built on the fly: h[oj] = sum_{t=0..4} dpad[k][oj + t].
    const int lane = tid & 31;
    const int wave = tid >> 5;
    const int n    = lane & 15;          // column within tile (B/D N index)
    const int m    = n;                  // A-matrix row (M index) for this lane
    const int kb   = (lane >> 4) << 1;   // K pair base: 0 (lanes 0-15) / 2 (16-31)
    const bool rowok = (oi0 + m) < OUT_H;

    // A fragments are loop/chunk invariant: banded box matrix,
    // W[m][k] = 1 iff k in [m, m+4]; invalid output rows -> 0.
    v2f afrag[5];
    #pragma unroll
    for (int c = 0; c < 5; ++c) {
        const int kg0 = 4 * c + kb;
        const int kg1 = kg0 + 1;
        afrag[c].x = (rowok && kg0 >= m && kg0 <= m + 4) ? 1.0f : 0.0f;
        afrag[c].y = (rowok && kg1 >= m && kg1 <= m + 4) ? 1.0f : 0.0f;
    }

    float lacc = 0.0f;
    for (int j = wave; j < N_CHUNK; j += 8) {
        const int col0 = j * 16;
        const int oj   = col0 + n;       // this lane's output column
        v8f acc = {0.f, 0.f, 0.f, 0.f, 0.f, 0.f, 0.f, 0.f};
        #pragma unroll
        for (int c = 0; c < 5; ++c) {
            const int r0 = 4 * c + kb;   // B rows {r0, r0+1} for this lane half
            const float* p0 = &dpad[r0][oj];
            const float* p1 = &dpad[r0 + 1][oj];
            v2f b;
            b.x = ((p0[0] + p0[1]) + (p0[2] + p0[3])) + p0[4];
            b.y = ((p1[0] + p1[1]) + (p1[2] + p1[3])) + p1[4];
            acc = __builtin_amdgcn_wmma_f32_16x16x4_f32(
                /*neg_a=*/false, afrag[c], /*neg_b=*/false, b,
                /*c_mod=*/(short)0, acc, /*reuse_a=*/false, /*reuse_b=*/false);
        }
        #pragma unroll
        for (int v = 0; v < 8; ++v) lacc += __builtin_fabsf(acc[v]);
    }

    // ---- Stage 3: block reduction, one partial per block ------------------
    red[tid] = lacc;
    __syncthreads();
    #pragma unroll
    for (int s = 128; s > 0; s >>= 1) {
        if (tid < s) red[tid] += red[tid + s];
        __syncthreads();
    }
    if (tid == 0) partials[img * ROW_TILES + rowTile] = red[0];
}

__global__ __launch_bounds__(256)
void boxdiff_reduce_kernel(const float* __restrict__ partials,
                           float* __restrict__ out) {
    __shared__ float red[256];
    const int tid = threadIdx.x;
    float s = 0.0f;
    for (int i = tid; i < NPART; i += 256) s += partials[i];
    red[tid] = s;
    __syncthreads();
    #pragma unroll
    for (int st = 128; st > 0; st >>= 1) {
        if (tid < st) red[tid] += red[tid + st];
        __syncthreads();
    }
    if (tid == 0) {
        // fold 1/25 box normalization and the mean over 64*516*516 outputs
        const double scale = 1.0 / (25.0 * (double)N_IMG * (double)OUT_H * (double)OUT_W);
        out[0] = (float)(red[0] * (float)scale);
    }
}

extern "C" void kernel_launch(void* const* d_in, const int* in_sizes, int n_in,
                              void* d_out, int out_size, void* d_ws, size_t ws_size,
                              hipStream_t stream) {
    (void)in_sizes; (void)n_in; (void)out_size; (void)ws_size;
    const float* x = (const float*)d_in[0];
    const float* y = (const float*)d_in[1];
    float* partials = (float*)d_ws;      // NPART floats of scratch
    float* out = (float*)d_out;

    dim3 grid(ROW_TILES, N_IMG);
    dim3 block(256);
    boxdiff_wmma_kernel<<<grid, block, 0, stream>>>(x, y, partials);
    boxdiff_reduce_kernel<<<1, 256, 0, stream>>>(partials, out);
}